// DepthwiseDeformConv1d_1537598292267
// MI455X (gfx1250) — compile-verified
//
#include <hip/hip_runtime.h>
#include <stdint.h>

typedef __attribute__((ext_vector_type(2))) float v2f;
typedef __attribute__((ext_vector_type(8))) float v8f;

#define C_CH  512
#define KTAP  7
#define T_IN  4096
#define T_OUT 4090
#define TILES 256          // ceil(4090/16); 256 % 8 waves == 0 -> uniform trip count

__global__ __launch_bounds__(256)
void deform_dwconv1d_kernel(const float* __restrict__ x,
                            const float* __restrict__ weight,
                            const float* __restrict__ offset_w,
                            const float* __restrict__ offset_b,
                            float* __restrict__ out)
{
    __shared__ __align__(16) float xs[T_IN + 32];   // x row + zero pad
    __shared__ float offw[56];                      // 49 used: offset_w[c, k, j]
    __shared__ float offb_s[8];                     // [7] = 0 (pad tap)
    __shared__ float wgt_s[8];                      // [7] = 0 (pad tap)
    __shared__ __align__(16) float wsc[8][16 * 8];  // per-wave offs scratch, t-major [t*8+k]

    const int bc   = blockIdx.x;                    // b*512 + c
    const int c    = bc & (C_CH - 1);
    const int tid  = threadIdx.x;
    const int lane = tid & 31;
    const int wid  = tid >> 5;

    const float* xrow = x   + (size_t)bc * T_IN;
    float*       orow = out + (size_t)bc * T_OUT;

    // ---- Phase 0: stage x row (16 KB) into LDS via CDNA5 async global->LDS DMA ----
    // low 32 bits of a generic pointer to LDS == LDS byte offset
    uint32_t xs_base = (uint32_t)(uintptr_t)(void*)xs;
    for (int i = tid; i < (T_IN * 4) / 16; i += 256) {   // 1024 x b128, 4 per thread
        uint32_t go = (uint32_t)i * 16u;                 // global byte offset (GVS mode)
        uint32_t lo = xs_base + go;                      // LDS byte address
        asm volatile("global_load_async_to_lds_b128 %0, %1, %2 offset:0"
                     :: "v"(lo), "v"(go), "s"(xrow) : "memory");
    }
    if (tid < 32) xs[T_IN + tid] = 0.0f;                 // zero pad tail
    if (tid < 49) offw[tid] = offset_w[c * 49 + tid];
    if (tid < 8) {
        offb_s[tid] = (tid < KTAP) ? offset_b[c * KTAP + tid] : 0.0f;
        wgt_s[tid]  = (tid < KTAP) ? weight[c * KTAP + tid]   : 0.0f;
    }
    asm volatile("s_wait_asynccnt 0" ::: "memory");
    __syncthreads();

    // ---- A fragments: 16x4 fp32, rows m=k (0..6 valid, 7..15 zero), K split j=0..3/4..6 ----
    const int m  = lane & 15;
    const int jb = (lane >> 4) << 1;                     // 0 or 2
    float a1x = 0.f, a1y = 0.f, a2x = 0.f, a2y = 0.f;
    if (m < KTAP) {
        a1x = offw[m * 7 + jb];
        a1y = offw[m * 7 + jb + 1];
        a2x = offw[m * 7 + 4 + jb];                      // j = 4 or 6
        if (jb == 0) a2y = offw[m * 7 + 5];              // j = 5; j = 7 -> 0
    }
    const v2f A1 = {a1x, a1y};
    const v2f A2 = {a2x, a2y};

    const int n     = lane & 15;                         // t within tile / B column
    const int kb    = (lane >> 4) << 1;                  // B K-rows: 0,1 / 2,3
    const int khalf = lane >> 4;
    const int kbase = khalf * 4;                         // k = 0..3 | 4..7 (k=7 is zero tap)

    for (int tile = wid; tile < TILES; tile += 8) {
        const int t0 = tile * 16;

        // ---- B fragments: 4x16 fp32 windows of x (rows K=j, cols N=t) ----
        v2f B1 = { xs[t0 + n + kb],     xs[t0 + n + kb + 1] };
        v2f B2 = { xs[t0 + n + 4 + kb], xs[t0 + n + 5 + kb] };

        // offs[k][t0+n] = sum_j W[k][j] * x[t0+n+j]   (two chained 16x16x4 WMMAs)
        v8f acc8 = {};
        acc8 = __builtin_amdgcn_wmma_f32_16x16x4_f32(false, A1, false, B1,
                                                     (short)0, acc8, false, false);
        acc8 = __builtin_amdgcn_wmma_f32_16x16x4_f32(false, A2, false, B2,
                                                     (short)0, acc8, false, false);

        // D layout: lanes 0-15, VGPR v -> row k=v, col t=lane. Spill rows 0..7
        // (row 7 is exactly 0) contiguously -> two ds_store_b128 per lane.
        if (lane < 16) {
            #pragma unroll
            for (int v = 0; v < 8; ++v)
                wsc[wid][lane * 8 + v] = acc8[v];
        }

        // ---- bilinear deformable sampling: uniform 4 taps/lane, k padded to 8 ----
        const int t = t0 + n;
        float acc = 0.0f;
        #pragma unroll
        for (int kk = 0; kk < 4; ++kk) {
            const int k = kbase + kk;
            const float off = wsc[wid][n * 8 + k] + offb_s[k];   // b128-gatherable
            const float pos = (float)(t + k) + off;
            const float fi  = floorf(pos);
            const float w1  = pos - fi;
            const int i0 = (int)fi;
            const int i1 = i0 + 1;
            int c0 = i0 < 0 ? 0 : (i0 > T_IN - 1 ? T_IN - 1 : i0);
            int c1 = i1 < 0 ? 0 : (i1 > T_IN - 1 ? T_IN - 1 : i1);
            float g0 = xs[c0]; if (i0 != c0) g0 = 0.0f;
            float g1 = xs[c1]; if (i1 != c1) g1 = 0.0f;
            acc += wgt_s[k] * (g0 * (1.0f - w1) + g1 * w1);
        }
        acc += __shfl_xor(acc, 16);                      // combine k halves (wave32)
        if (khalf == 0 && t < T_OUT) orow[t] = acc;
    }
}

extern "C" void kernel_launch(void* const* d_in, const int* in_sizes, int n_in,
                              void* d_out, int out_size, void* d_ws, size_t ws_size,
                              hipStream_t stream) {
    const float* x        = (const float*)d_in[0];
    const float* weight   = (const float*)d_in[1];
    const float* offset_w = (const float*)d_in[2];
    const float* offset_b = (const float*)d_in[3];
    float* out = (float*)d_out;

    const int B = in_sizes[0] / (C_CH * T_IN);           // 8
    dim3 grid(B * C_CH), block(256);
    deform_dwconv1d_kernel<<<grid, block, 0, stream>>>(x, weight, offset_w, offset_b, out);
}